// ContextualAttention_39015482917457
// MI455X (gfx1250) — compile-verified
//
#include <hip/hip_runtime.h>
#include <hip/hip_bf16.h>

// ---------------------------------------------------------------------------
// Contextual attention, MI455X (gfx1250, wave32, WMMA-f32).
//   B=2, C=128, H=W=128, rate=2 -> 64x64 grids, L=4096 patches.
//   Score conv decomposed as GEMM (4096x128x4096) + 9-tap diagonal stencil.
//   Deconv as GEMM (4096x4096x2048) + stride-2 overlap-add gather.
//   GEMM: 128x128 block tile, K-chunk 32, 8 waves x (4x2) 16x16 subtiles,
//         V_WMMA_F32_16X16X4_F32 (fp32 fidelity for the x10 softmax).
// ---------------------------------------------------------------------------

#define HS   64            // downsampled spatial
#define LQ   4096          // HS*HS patches / locations
#define CCH  128           // channels
#define HFULL 128          // full-res spatial
#define PN   2048          // CCH * 16 (c,i,j) columns of Braw / Out2

typedef float v2f __attribute__((ext_vector_type(2)));
typedef float v8f __attribute__((ext_vector_type(8)));

// ------------------------- staging: f_ds / b_ds ----------------------------
// Fp[p][c]  (row-major, p=hp*64+wp)   Bq[c][q]  (row-major)
__global__ __launch_bounds__(256) void ca_stage(const float* __restrict__ f,
                                                const float* __restrict__ b,
                                                float* __restrict__ Fp,
                                                float* __restrict__ Bq) {
  int idx = blockIdx.x * 256 + threadIdx.x;          // 0 .. 2*LQ*CCH-1
  if (idx < LQ * CCH) {
    int p = idx >> 7, c = idx & 127;
    int hp = p >> 6, wp = p & 63;
    Fp[idx] = f[((size_t)c * HFULL + 2 * hp) * HFULL + 2 * wp];
  } else {
    int j = idx - LQ * CCH;
    int c = j >> 12, q = j & 4095;
    int hq = q >> 6, wq = q & 63;
    Bq[(size_t)c * LQ + q] = b[((size_t)c * HFULL + 2 * hq) * HFULL + 2 * wq];
  }
}

// invnorm[q] = rsqrt( sum_{c,3x3} b_ds^2 + 1152*1e-4 )   (eps summed per elem)
__global__ __launch_bounds__(256) void ca_invnorm(const float* __restrict__ Bq,
                                                  float* __restrict__ invn) {
  int q = blockIdx.x * 256 + threadIdx.x;
  int hq = q >> 6, wq = q & 63;
  float s = 0.f;
  for (int c = 0; c < CCH; ++c) {
    const float* base = Bq + (size_t)c * LQ;
    for (int dy = -1; dy <= 1; ++dy) {
      int hh = hq + dy; if ((unsigned)hh >= HS) continue;
      for (int dx = -1; dx <= 1; ++dx) {
        int ww = wq + dx; if ((unsigned)ww >= HS) continue;
        float v = base[hh * HS + ww];
        s += v * v;
      }
    }
  }
  invn[q] = rsqrtf(s + 1152.0f * 1e-4f);   // jnp.sum(w*w + 1e-4) over 1152 elems
}

// mm[q] = (mean of 3x3 patch of mask_ds == 0) ? 1 : 0
__global__ __launch_bounds__(256) void ca_mm(const float* __restrict__ mask,
                                             float* __restrict__ mmv) {
  int q = blockIdx.x * 256 + threadIdx.x;
  int hq = q >> 6, wq = q & 63;
  float s = 0.f;
  for (int dy = -1; dy <= 1; ++dy) {
    int hh = hq + dy; if ((unsigned)hh >= HS) continue;
    for (int dx = -1; dx <= 1; ++dx) {
      int ww = wq + dx; if ((unsigned)ww >= HS) continue;
      s += mask[(2 * hh) * HFULL + 2 * ww];
    }
  }
  mmv[q] = (s == 0.f) ? 1.f : 0.f;
}

// Braw[q][(c,i,j)] = b[c][2hq+i-1][2wq+j-1]  (zero OOB), 4x4 raw patches
__global__ __launch_bounds__(256) void ca_braw(const float* __restrict__ b,
                                               float* __restrict__ Braw) {
  size_t idx = (size_t)blockIdx.x * 256 + threadIdx.x; // LQ*PN
  int n = (int)(idx & (PN - 1));
  int q = (int)(idx >> 11);
  int c = n >> 4, i = (n >> 2) & 3, j = n & 3;
  int hq = q >> 6, wq = q & 63;
  int u = 2 * hq + i - 1, v = 2 * wq + j - 1;
  float val = 0.f;
  if ((unsigned)u < HFULL && (unsigned)v < HFULL)
    val = b[((size_t)c * HFULL + u) * HFULL + v];
  Braw[idx] = val;
}

// -------------------- generic WMMA-f32 GEMM  C = A * B ---------------------
// A [M][K], B [K][N], C [M][N] row-major. M,N multiples of 128; K multiple of 32.
// 256 threads = 8 waves (2x4 grid). 128x128 C tile; wave owns 4x2 16x16 subtiles.
#define KCH 32
__global__ __launch_bounds__(256) void ca_gemm_f32(const float* __restrict__ A,
                                                   const float* __restrict__ B,
                                                   float* __restrict__ C,
                                                   int M, int N, int K) {
  __shared__ __attribute__((aligned(16))) float As[128][36];  // stride 36: banks 4*(9r mod 16)
  __shared__ __attribute__((aligned(16))) float Bs[KCH][132];
  const int n0 = blockIdx.x * 128;
  const int m0 = blockIdx.y * 128;
  const int t = threadIdx.x;
  const int wave = t >> 5;
  const int lane = t & 31;
  const int half = lane >> 4;     // lane half selects K offset (+2) per ISA layout
  const int l16  = lane & 15;

  const int wr = wave >> 2;       // 2 wave-rows x 4 wave-cols
  const int wc = wave & 3;

  v8f acc[4][2];
#pragma unroll
  for (int am = 0; am < 4; ++am)
#pragma unroll
    for (int bn = 0; bn < 2; ++bn)
      acc[am][bn] = (v8f){0.f,0.f,0.f,0.f,0.f,0.f,0.f,0.f};

  for (int kc = 0; kc < K; kc += KCH) {
    // load A chunk: 128x32 floats = 1024 float4, 4 per thread
#pragma unroll
    for (int i = 0; i < 4; ++i) {
      int idx = t + 256 * i;                 // 0..1023
      int r = idx >> 3, c4 = idx & 7;
      float4 av = *reinterpret_cast<const float4*>(
          &A[(size_t)(m0 + r) * K + kc + c4 * 4]);
      *reinterpret_cast<float4*>(&As[r][c4 * 4]) = av;
    }
    // load B chunk: 32x128 floats = 1024 float4, 4 per thread
#pragma unroll
    for (int i = 0; i < 4; ++i) {
      int idx = t + 256 * i;
      int r = idx >> 5, c4 = idx & 31;
      float4 bv = *reinterpret_cast<const float4*>(
          &B[(size_t)(kc + r) * N + n0 + c4 * 4]);
      *reinterpret_cast<float4*>(&Bs[r][c4 * 4]) = bv;
    }
    if (kc + KCH < K) {                      // speculative L2 prefetch of next chunk
      __builtin_prefetch(&A[(size_t)(m0 + (t >> 1)) * K + kc + KCH], 0, 1);
      __builtin_prefetch(&B[(size_t)(kc + KCH + (t >> 3)) * N + n0], 0, 1);
    }
    __syncthreads();
#pragma unroll
    for (int k = 0; k < KCH; k += 4) {
      const int ka = k + 2 * half;
      // A 16x4 fragments: lanes 0-15 rows, VGPR0/1 = K{0,1}(+2 for lane half 1)
      v2f a[4], bb[2];
#pragma unroll
      for (int am = 0; am < 4; ++am) {
        int row = wr * 16 + am * 32 + l16;
        a[am].x = As[row][ka];
        a[am].y = As[row][ka + 1];
      }
      // B 4x16 fragments (mirrored K mapping, N striped across lanes)
#pragma unroll
      for (int bn = 0; bn < 2; ++bn) {
        int col = wc * 16 + bn * 64 + l16;
        bb[bn].x = Bs[ka][col];
        bb[bn].y = Bs[ka + 1][col];
      }
#pragma unroll
      for (int am = 0; am < 4; ++am)
#pragma unroll
        for (int bn = 0; bn < 2; ++bn)
          acc[am][bn] = __builtin_amdgcn_wmma_f32_16x16x4_f32(
              false, a[am], false, bb[bn], (short)0, acc[am][bn], false, false);
    }
    __syncthreads();
  }
  // C/D layout: VGPR r -> row r (lanes 0-15) / row r+8 (lanes 16-31), col = lane%16
#pragma unroll
  for (int am = 0; am < 4; ++am) {
    int mrow = m0 + wr * 16 + am * 32 + 8 * half;
#pragma unroll
    for (int bn = 0; bn < 2; ++bn) {
      int ncol = n0 + wc * 16 + bn * 64 + l16;
#pragma unroll
      for (int r = 0; r < 8; ++r)
        C[(size_t)(mrow + r) * N + ncol] = acc[am][bn][r];
    }
  }
}

// S[p][q] = invn[q] * sum_{dy,dx in 3x3} G[p+d][q+d]   (2D bounds on both)
__global__ __launch_bounds__(256) void ca_window(const float* __restrict__ G,
                                                 const float* __restrict__ invn,
                                                 float* __restrict__ S) {
  size_t idx = (size_t)blockIdx.x * 256 + threadIdx.x;  // LQ*LQ, q fastest
  int q = (int)(idx & (LQ - 1));
  int p = (int)(idx >> 12);
  int hp = p >> 6, wp = p & 63, hq = q >> 6, wq = q & 63;
  float s = 0.f;
  for (int dy = -1; dy <= 1; ++dy) {
    int hpp = hp + dy, hqq = hq + dy;
    if ((unsigned)hpp >= HS || (unsigned)hqq >= HS) continue;
    for (int dx = -1; dx <= 1; ++dx) {
      int wpp = wp + dx, wqq = wq + dx;
      if ((unsigned)wpp >= HS || (unsigned)wqq >= HS) continue;
      s += G[(size_t)(hpp * HS + wpp) * LQ + (hqq * HS + wqq)];
    }
  }
  S[idx] = s * invn[q];
}

// fuse conv #1: diagonal 3-tap over flat (q,p); symmetric in (p,q) storage
__global__ __launch_bounds__(256) void ca_fuse1(const float* __restrict__ S,
                                                float* __restrict__ Y) {
  size_t idx = (size_t)blockIdx.x * 256 + threadIdx.x;
  int q = (int)(idx & (LQ - 1));
  int p = (int)(idx >> 12);
  float s = 0.f;
  for (int e = -1; e <= 1; ++e) {
    int pp = p + e, qq = q + e;
    if ((unsigned)pp < LQ && (unsigned)qq < LQ)
      s += S[(size_t)pp * LQ + qq];
  }
  Y[idx] = s;
}

// fuse conv #2: diagonal 3-tap over transposed-flat indices (wrap across columns)
__global__ __launch_bounds__(256) void ca_fuse2(const float* __restrict__ Y,
                                                float* __restrict__ Z) {
  size_t idx = (size_t)blockIdx.x * 256 + threadIdx.x;
  int q = (int)(idx & (LQ - 1));
  int p = (int)(idx >> 12);
  int ap = (p & 63) * 64 + (p >> 6);   // transposed flat
  int aq = (q & 63) * 64 + (q >> 6);
  float s = 0.f;
  for (int e = -1; e <= 1; ++e) {
    int a1 = ap + e, a2 = aq + e;
    if ((unsigned)a1 < LQ && (unsigned)a2 < LQ) {
      int pp = (a1 & 63) * 64 + (a1 >> 6);
      int qq = (a2 & 63) * 64 + (a2 >> 6);
      s += Y[(size_t)pp * LQ + qq];
    }
  }
  Z[idx] = s;
}

// row-wise softmax over q (contiguous): At[p][q] = softmax_q(10*Z*mm) * mm
__global__ __launch_bounds__(256) void ca_softmax(const float* __restrict__ Z,
                                                  const float* __restrict__ mmv,
                                                  float* __restrict__ At) {
  __shared__ float red[256];
  const int p = blockIdx.x;
  const int t = threadIdx.x;
  const float* row = Z + (size_t)p * LQ;
  float vals[16];
  float m = -3.4e38f;
#pragma unroll
  for (int i = 0; i < 16; ++i) {
    int q = t + 256 * i;
    float l = 10.f * row[q] * mmv[q];
    vals[i] = l;
    m = fmaxf(m, l);
  }
  red[t] = m; __syncthreads();
  for (int s = 128; s > 0; s >>= 1) {
    if (t < s) red[t] = fmaxf(red[t], red[t + s]);
    __syncthreads();
  }
  m = red[0]; __syncthreads();
  float sum = 0.f;
#pragma unroll
  for (int i = 0; i < 16; ++i) { vals[i] = __expf(vals[i] - m); sum += vals[i]; }
  red[t] = sum; __syncthreads();
  for (int s = 128; s > 0; s >>= 1) {
    if (t < s) red[t] += red[t + s];
    __syncthreads();
  }
  float inv = 1.f / red[0];
  float* orow = At + (size_t)p * LQ;
#pragma unroll
  for (int i = 0; i < 16; ++i) {
    int q = t + 256 * i;
    orow[q] = vals[i] * inv * mmv[q];
  }
}

// overlap-add: out[c][x][y] = 0.25 * sum over <=4 taps of Out2[(hf,wf)][(c,i,j)]
__global__ __launch_bounds__(256) void ca_scatter(const float* __restrict__ Out2,
                                                  float* __restrict__ out) {
  size_t idx = (size_t)blockIdx.x * 256 + threadIdx.x;  // CCH*HFULL*HFULL
  int y = (int)(idx & 127);
  int x = (int)((idx >> 7) & 127);
  int c = (int)(idx >> 14);
  int ipar = (x + 1) & 1, jpar = (y + 1) & 1;
  float s = 0.f;
  for (int di = 0; di < 2; ++di) {
    int i = ipar + 2 * di;
    int num = x + 1 - i;
    if (num < 0) continue;
    int hf = num >> 1; if (hf >= HS) continue;
    for (int dj = 0; dj < 2; ++dj) {
      int j = jpar + 2 * dj;
      int num2 = y + 1 - j;
      if (num2 < 0) continue;
      int wf = num2 >> 1; if (wf >= HS) continue;
      s += Out2[(size_t)(hf * HS + wf) * PN + (c << 4) + (i << 2) + j];
    }
  }
  out[idx] = 0.25f * s;
}

// ---------------------------------------------------------------------------
extern "C" void kernel_launch(void* const* d_in, const int* in_sizes, int n_in,
                              void* d_out, int out_size, void* d_ws, size_t ws_size,
                              hipStream_t stream) {
  const float* f    = (const float*)d_in[0];   // (2,128,128,128)
  const float* b    = (const float*)d_in[1];   // (2,128,128,128)
  const float* mask = (const float*)d_in[2];   // (1,1,128,128)
  float* out = (float*)d_out;                  // (2,128,128,128)

  char* ws = (char*)d_ws;
  const size_t MB = (size_t)1 << 20;
  float* W0   = (float*)(ws);                      // 64 MB: G -> Y -> At
  float* W1   = (float*)(ws + 64 * MB);            // 64 MB: S -> Z -> Out2
  float* W2   = (float*)(ws + 128 * MB);           // 32 MB: Braw
  float* Fp   = (float*)(ws + 160 * MB);           // 2 MB
  float* Bq   = Fp + (size_t)LQ * CCH;             // 2 MB
  float* invn = Bq + (size_t)CCH * LQ;             // 16 KB
  float* mmv  = invn + LQ;                         // 16 KB

  // mask gate (batch independent)
  ca_mm<<<LQ / 256, 256, 0, stream>>>(mask, mmv);

  for (int bt = 0; bt < 2; ++bt) {
    const float* fb = f + (size_t)bt * CCH * HFULL * HFULL;
    const float* bb = b + (size_t)bt * CCH * HFULL * HFULL;
    float* ob = out + (size_t)bt * CCH * HFULL * HFULL;

    ca_stage<<<(2 * LQ * CCH) / 256, 256, 0, stream>>>(fb, bb, Fp, Bq);
    ca_invnorm<<<LQ / 256, 256, 0, stream>>>(Bq, invn);
    ca_braw<<<((size_t)LQ * PN) / 256, 256, 0, stream>>>(bb, W2);

    // G[p][q] = sum_c f_ds[c,p] * b_ds[c,q]   (M=4096, N=4096, K=128)
    ca_gemm_f32<<<dim3(LQ / 128, LQ / 128), 256, 0, stream>>>(Fp, Bq, W0, LQ, LQ, CCH);

    ca_window<<<((size_t)LQ * LQ) / 256, 256, 0, stream>>>(W0, invn, W1);
    ca_fuse1<<<((size_t)LQ * LQ) / 256, 256, 0, stream>>>(W1, W0);
    ca_fuse2<<<((size_t)LQ * LQ) / 256, 256, 0, stream>>>(W0, W1);
    ca_softmax<<<LQ, 256, 0, stream>>>(W1, mmv, W0);

    // Out2[p][n] = sum_q At[p][q] * Braw[q][n]  (M=4096, N=2048, K=4096)
    ca_gemm_f32<<<dim3(PN / 128, LQ / 128), 256, 0, stream>>>(W0, W2, W1, LQ, PN, LQ);

    ca_scatter<<<((size_t)CCH * HFULL * HFULL) / 256, 256, 0, stream>>>(W1, ob);
  }
}